// QuantumLayer_36610301231212
// MI455X (gfx1250) — compile-verified
//
#include <hip/hip_runtime.h>
#include <math.h>

// ---- static problem configuration (matches reference) ----
#define MM        10      // modes
#define NPH       5       // photons
#define BATCH     4096
#define OUTF      64
#define KDIM      2002    // states with 5 photons in 10 modes
#define PSTRIDE   2016    // padded probs row stride (floats)

// number of Fock states: Cnt(m modes, n photons) = C(m+n-1, n), m=1..10, n=0..5
__device__ __constant__ int d_CNT[10][6] = {
    {1, 1,  1,  1,   1,   1},
    {1, 2,  3,  4,   5,   6},
    {1, 3,  6, 10,  15,  21},
    {1, 4, 10, 20,  35,  56},
    {1, 5, 15, 35,  70, 126},
    {1, 6, 21, 56, 126, 252},
    {1, 7, 28, 84, 210, 462},
    {1, 8, 36, 120, 330, 792},
    {1, 9, 45, 165, 495, 1287},
    {1, 10, 55, 220, 715, 2002}};

__device__ __constant__ int d_NS[6]  = {1, 10, 55, 220, 715, 2002};
__device__ __constant__ int d_OFF[5] = {0, 10, 65, 285, 1000};   // table row offsets per level

typedef __attribute__((ext_vector_type(2))) float v2f;
typedef __attribute__((ext_vector_type(8))) float v8f;

// ---------- combinatorics: rank/unrank in the reference's _enum ordering ----------
// _enum(m,n): first mode count descending, recurse on remaining modes.
__device__ inline void decodeState(int r, int n, int* s) {
    int rem = n;
    for (int i = 0; i < 9; ++i) {
        int mm = 10 - i;           // modes remaining (incl. i)
        int chosen = 0;
        for (int k = rem; k >= 0; --k) {
            int c = d_CNT[mm - 2][rem - k];  // Cnt(mm-1, rem-k)
            if (r < c) { chosen = k; break; }
            r -= c;
        }
        s[i] = chosen;
        rem -= chosen;
    }
    s[9] = rem;
}

__device__ inline int rankState(const int* s) {
    int rem = 0;
    for (int i = 0; i < 10; ++i) rem += s[i];
    int r = 0;
    for (int i = 0; i < 9; ++i) {
        int mm = 10 - i;
        for (int k = rem; k > s[i]; --k) r += d_CNT[mm - 2][rem - k];
        rem -= s[i];
    }
    return r;
}

// ---------- kernel 1: build SLOS transition tables (idx, sqrt) into ws ----------
// Row t in [0,3002): flattened over levels k=0..4, entries for states of level k+1.
__global__ void buildTables(int2* __restrict__ tab) {
    int t = blockIdx.x * blockDim.x + threadIdx.x;
    if (t >= 3002) return;
    int k = 0;
    while (k < 4 && t >= d_OFF[k] + d_NS[k + 1]) ++k;
    int r = t - d_OFF[k];
    int s[10];
    decodeState(r, k + 1, s);
    int2* row = tab + (size_t)t * 10;
    for (int j = 0; j < 10; ++j) {
        int idx = 0;
        float sq = 0.0f;
        if (s[j] > 0) {
            int p[10];
            for (int q = 0; q < 10; ++q) p[q] = s[q];
            p[j] -= 1;
            idx = rankState(p);
            sq = sqrtf((float)s[j]);
        }
        row[j] = make_int2(idx, __float_as_int(sq));
    }
}

// ---------- kernel 2: build batch-independent V = BS9 D(th8) ... BS1 D(th0) BS0 ----------
// Column-parallel: thread n evolves column n. Stored column-major: wsV[n*10 + j] = V[j][n].
__global__ void buildV(const float* __restrict__ theta, float2* __restrict__ wsV) {
    int n = threadIdx.x;
    if (n >= 10) return;
    const float inv = 0.70710678118654752440f;
    float2 col[10];
    for (int j = 0; j < 10; ++j) col[j] = make_float2(0.f, 0.f);
    // col = BS0 * e_n : pairs (p,p+1), offset 0
    col[n]     = make_float2(inv, 0.f);
    col[n ^ 1] = make_float2(0.f, inv);
    for (int l = 1; l < 10; ++l) {
        // per-mode phase shifters (row scaling)
        for (int j = 0; j < 10; ++j) {
            float s, c;
            sincosf(theta[(l - 1) * 10 + j], &s, &c);
            float2 a = col[j];
            col[j] = make_float2(c * a.x - s * a.y, c * a.y + s * a.x);
        }
        // fixed 50:50 BS layer, alternating Clements offsets
        int offs = l & 1;
        for (int p = offs; p + 1 < 10; p += 2) {
            float2 a = col[p], b = col[p + 1];
            col[p]     = make_float2(inv * (a.x - b.y), inv * (a.y + b.x));
            col[p + 1] = make_float2(inv * (b.x - a.y), inv * (b.y + a.x));
        }
    }
    for (int j = 0; j < 10; ++j) wsV[n * 10 + j] = col[j];
}

// ---------- kernel 3: SLOS per batch element; write |amp5|^2 row to ws ----------
__global__ __launch_bounds__(128) void slosKernel(const float* __restrict__ x,
                                                  const float2* __restrict__ wsV,
                                                  const int2* __restrict__ tab,
                                                  float* __restrict__ probs) {
    __shared__ float2 colS[5][10];   // U_b[:,k] = V[:,k] * e^{i 2pi x[b,k]}
    __shared__ float2 ampEven[715];  // levels 0, 2, 4
    __shared__ float2 ampOdd[220];   // levels 1, 3
    const int b   = blockIdx.x;
    const int tid = threadIdx.x;
    if (tid < 50) {
        int k = tid / 10, j = tid % 10;
        float s, c;
        sincosf(6.283185307179586f * x[b * 10 + k], &s, &c);
        float2 v = wsV[k * 10 + j];
        colS[k][j] = make_float2(c * v.x - s * v.y, c * v.y + s * v.x);
    }
    if (tid == 0) ampEven[0] = make_float2(1.f, 0.f);  // vacuum
    __syncthreads();

    float* probsRow = probs + (size_t)b * PSTRIDE;
    for (int k = 0; k < 5; ++k) {
        const float2* src = (k & 1) ? ampOdd : ampEven;
        float2* dst       = (k & 1) ? ampEven : ampOdd;
        const int nOut = d_NS[k + 1];
        const int2* trow = tab + (size_t)d_OFF[k] * 10;
        for (int t = tid; t < nOut; t += 128) {
            const int2* e = trow + t * 10;
            float ax = 0.f, ay = 0.f;
#pragma unroll
            for (int j = 0; j < 10; ++j) {
                int2 ev = e[j];
                float sq = __int_as_float(ev.y);
                float2 a = src[ev.x];
                float2 c = colS[k][j];
                float rx = a.x * sq, ry = a.y * sq;
                ax = fmaf(c.x, rx, fmaf(-c.y, ry, ax));
                ay = fmaf(c.x, ry, fmaf(c.y, rx, ay));
            }
            if (k < 4) dst[t] = make_float2(ax, ay);
            else       probsRow[t] = ax * ax + ay * ay;  // |amp|^2 (prod s_i! == 1)
        }
        __syncthreads();
    }
}

// ---------- kernel 4: out = probs @ W + b via V_WMMA_F32_16X16X4_F32 ----------
// 256 blocks x 128 threads: 16 batch rows/block, 4 waves -> one 16-col N tile each.
__global__ __launch_bounds__(128) void gemmKernel(const float* __restrict__ probs,
                                                  const float* __restrict__ W,
                                                  const float* __restrict__ bias,
                                                  float* __restrict__ out) {
    const int lane = threadIdx.x & 31;
    const int wave = threadIdx.x >> 5;   // 0..3 -> N tile
    const int row0 = blockIdx.x * 16;
    const int ar   = lane & 15;          // A-row / B-col within tile
    const int half = lane >> 4;          // K-half select per WMMA layout
    const int colN = wave * 16 + ar;

    const float* Arow = probs + (size_t)(row0 + ar) * PSTRIDE + 2 * half;
    const float* Bp   = W + (size_t)(2 * half) * OUTF + colN;

    v8f acc = {0.f, 0.f, 0.f, 0.f, 0.f, 0.f, 0.f, 0.f};
#pragma unroll 4
    for (int kk = 0; kk < 2000; kk += 4) {
        v2f a;  a.x = Arow[kk];               a.y = Arow[kk + 1];
        v2f bb; bb.x = Bp[(size_t)kk * OUTF]; bb.y = Bp[(size_t)kk * OUTF + OUTF];
        acc = __builtin_amdgcn_wmma_f32_16x16x4_f32(false, a, false, bb,
                                                    (short)0, acc, false, false);
    }
    // K tail (k = 2000, 2001) on VALU + bias + store (D layout: vgpr v -> M = v + 8*half)
    const float w0 = W[(size_t)2000 * OUTF + colN];
    const float w1 = W[(size_t)2001 * OUTF + colN];
    const float bc = bias[colN];
    const int mBase = half * 8;
#pragma unroll
    for (int v = 0; v < 8; ++v) {
        int m = row0 + mBase + v;
        const float* Ar = probs + (size_t)m * PSTRIDE;
        float sum = acc[v] + Ar[2000] * w0 + Ar[2001] * w1 + bc;
        out[(size_t)m * OUTF + colN] = sum;
    }
}

// ---------- workspace layout ----------
//   [0, 800)                 : V (100 float2)
//   [1024, 1024+240160)      : transition tables (3002*10 int2)
//   [241664, +4096*2016*4)   : probs rows, stride-padded
static const size_t WS_OFF_V     = 0;
static const size_t WS_OFF_TAB   = 1024;
static const size_t WS_OFF_PROBS = 241664;

extern "C" void kernel_launch(void* const* d_in, const int* in_sizes, int n_in,
                              void* d_out, int out_size, void* d_ws, size_t ws_size,
                              hipStream_t stream) {
    const float* x     = (const float*)d_in[0];   // [4096, 10]
    const float* theta = (const float*)d_in[1];   // [90]
    const float* W     = (const float*)d_in[2];   // [2002, 64]
    const float* bias  = (const float*)d_in[3];   // [64]
    float* out = (float*)d_out;                   // [4096, 64]

    char* ws = (char*)d_ws;
    float2* wsV    = (float2*)(ws + WS_OFF_V);
    int2*   wsTab  = (int2*)(ws + WS_OFF_TAB);
    float*  wsProb = (float*)(ws + WS_OFF_PROBS);

    buildTables<<<24, 128, 0, stream>>>(wsTab);
    buildV<<<1, 32, 0, stream>>>(theta, wsV);
    slosKernel<<<BATCH, 128, 0, stream>>>(x, wsV, wsTab, wsProb);
    gemmKernel<<<BATCH / 16, 128, 0, stream>>>(wsProb, W, bias, out);
}